// DecoderLSTM_17746804867665
// MI455X (gfx1250) — compile-verified
//
#include <hip/hip_runtime.h>
#include <hip/hip_bf16.h>
#include <math.h>

// ---------------------------------------------------------------------------
// DecoderLSTM for MI455X (gfx1250, wave32).
//   B=64, T=32, X=512, HID=512, VOCAB=10000, all fp32.
// - All GEMMs on v_wmma_f32_16x16x4_f32 (exact fp32, matrix pipes).
// - The dominant data movement (W_lin: 20.5MB/step, 655MB total, HBM floor
//   ~28us @ 23.3TB/s) is streamed by the Tensor Data Mover into LDS with
//   double buffering (tensor_load_to_lds + s_wait_tensorcnt), and consumed
//   as WMMA B-fragments via ds_load.
// ---------------------------------------------------------------------------

#define Bsz   64
#define Tsz   32
#define Xdim  512
#define Hdim  512
#define Vsz   10000

typedef __attribute__((ext_vector_type(2))) float v2f;
typedef __attribute__((ext_vector_type(8))) float v8f;
typedef __attribute__((ext_vector_type(4))) unsigned int u32x4;
typedef __attribute__((ext_vector_type(8))) int i32x8;
typedef __attribute__((ext_vector_type(4))) int i32x4;

#if defined(__has_builtin)
#if __has_builtin(__builtin_amdgcn_wmma_f32_16x16x4_f32)
#define HAVE_WMMA_F32X4 1
#endif
#if __has_builtin(__builtin_amdgcn_tensor_load_to_lds)
#define HAVE_TDM 1
#endif
#endif

// D(16x16,f32) += A(16x4,f32) * B(4x16,f32), one wave (32 lanes).
__device__ __forceinline__ v8f wmma4(v2f a, v2f b, v8f c) {
#if defined(HAVE_WMMA_F32X4)
  return __builtin_amdgcn_wmma_f32_16x16x4_f32(
      false, a, false, b, (short)0, c, false, false);
#else
  #pragma unroll
  for (int r = 0; r < 8; ++r) c[r] += a.x * b.x + a.y * b.y;
  return c;
#endif
}

// A fragment: 16x4 slice of row-major A[ldA]; lanes 0-15: K={k,k+1},
// lanes 16-31: K={k+2,k+3}; M = lane&15.
__device__ __forceinline__ v2f load_a(const float* __restrict__ A, int lda,
                                      int m0, int k, int lane) {
  int m  = m0 + (lane & 15);
  int kk = k + ((lane >> 4) << 1);
  const float* p = A + (size_t)m * lda + kk;
  v2f a; a.x = p[0]; a.y = p[1];
  return a;
}

// B fragment: 4x16 slice of row-major B[ldB] (global).
__device__ __forceinline__ v2f load_b(const float* __restrict__ B, int ldb,
                                      int k, int n0, int lane) {
  int n  = n0 + (lane & 15);
  int kk = k + ((lane >> 4) << 1);
  const float* p = B + (size_t)kk * ldb + n;
  v2f b; b.x = p[0]; b.y = p[ldb];
  return b;
}

// D store: VGPR r -> rows m0+r (lanes 0-15) / m0+8+r (lanes 16-31).
__device__ __forceinline__ void store_d(float* __restrict__ D, int ldd,
                                        int m0, int n0, int lane, v8f c,
                                        const float* __restrict__ bias) {
  int col   = n0 + (lane & 15);
  int rbase = m0 + ((lane >> 4) << 3);
  float bv = bias ? bias[col] : 0.0f;
  #pragma unroll
  for (int r = 0; r < 8; ++r)
    D[(size_t)(rbase + r) * ldd + col] = c[r] + bv;
}

// ---------------------------------------------------------------------------
// TDM: async 2D tile load (rows x cols of fp32) from global into LDS.
// D# built per CDNA5 ISA (group0 + group1; groups 2/3/4
// zero => 2D tensor). Issued by one wave; tracked by TENSORcnt.
// clang-23 toolchain: 6-arg builtin (g0 u32x4, g1 i32x8, g2 i32x4, g3 i32x4,
// g4 i32x8, i32 cpol).
// ---------------------------------------------------------------------------
__device__ __forceinline__ void tdm_load_2d(unsigned int lds_addr,
                                            const void* gaddr,
                                            unsigned int dim0,      // tensor cols remaining
                                            unsigned int dim1,      // tensor rows remaining
                                            unsigned int stride0,   // row stride (elems)
                                            unsigned int tile0,     // tile cols
                                            unsigned int tile1) {   // tile rows
#if defined(HAVE_TDM)
  unsigned long long ga = (unsigned long long)(uintptr_t)gaddr;
  u32x4 g0;
  g0[0] = 1u;                                   // count=1, user descriptor
  g0[1] = lds_addr;                             // LDS byte address
  g0[2] = (unsigned int)ga;                     // global_addr[31:0]
  g0[3] = (unsigned int)((ga >> 32) & 0x01FFFFFFu) | 0x80000000u; // addr[56:32] | type=2
  i32x8 g1;
  g1[0] = (int)(2u << 16);                      // wg_mask=0, data_size=2 (4B)
  g1[1] = (int)((dim0 & 0xFFFFu) << 16);        // tensor_dim0[15:0] (atomic_barrier=0)
  g1[2] = (int)(((dim0 >> 16) & 0xFFFFu) | ((dim1 & 0xFFFFu) << 16));
  g1[3] = (int)(((dim1 >> 16) & 0xFFFFu) | ((tile0 & 0xFFFFu) << 16));
  g1[4] = (int)(tile1 & 0xFFFFu);               // tile_dim1; tile_dim2=0 (2D)
  g1[5] = (int)stride0;                         // tensor_dim0_stride[31:0]
  g1[6] = 0;                                    // stride0 hi | dim1_stride lo
  g1[7] = 0;
  i32x4 g2 = {0, 0, 0, 0};
  i32x4 g3 = {0, 0, 0, 0};
  i32x8 g4 = {0, 0, 0, 0, 0, 0, 0, 0};
  __builtin_amdgcn_tensor_load_to_lds(g0, g1, g2, g3, g4, 0);
#else
  (void)lds_addr; (void)gaddr; (void)dim0; (void)dim1;
  (void)stride0; (void)tile0; (void)tile1;
#endif
}

struct GateParams {
  const float* Wh[4];   // [HID, HID], gate order i,f,o,z
  const float* Wx[4];   // [X,   HID]
  const float* bias[4]; // [HID]
};

// ---------------------------------------------------------------------------
// Kernel 1: gate pre-activations  G[g] = x@Wx[g] + h@Wh[g] + b[g]
// grid = 4 gates * 8 col-blocks of 64 = 32 blocks; 8 waves each, 2 tiles/wave.
// Weights are L2-resident across all 32 timesteps (8MB total).
// ---------------------------------------------------------------------------
__global__ void __launch_bounds__(256)
lstm_gates_wmma(const float* __restrict__ xA, int ldx,
                const float* __restrict__ hA, int ldh,
                GateParams gp, float* __restrict__ gates /* [4][64][512] */) {
  const int g    = blockIdx.x & 3;
  const int nblk = blockIdx.x >> 2;       // 0..7 -> 64 columns each
  const int wave = threadIdx.x >> 5;      // 0..7
  const int lane = threadIdx.x & 31;

  const float* Wx = gp.Wx[g];
  const float* Wh = gp.Wh[g];
  const float* bv = gp.bias[g];
  float* Dg = gates + (size_t)g * Bsz * Hdim;

  for (int tsel = wave; tsel < 16; tsel += 8) {   // 4 m-tiles x 4 n-tiles
    const int m0 = (tsel & 3) * 16;
    const int n0 = nblk * 64 + (tsel >> 2) * 16;
    v8f acc = {};
    #pragma unroll 8
    for (int k = 0; k < Xdim; k += 4)
      acc = wmma4(load_a(xA, ldx, m0, k, lane), load_b(Wx, Hdim, k, n0, lane), acc);
    #pragma unroll 8
    for (int k = 0; k < Hdim; k += 4)
      acc = wmma4(load_a(hA, ldh, m0, k, lane), load_b(Wh, Hdim, k, n0, lane), acc);
    store_d(Dg, Hdim, m0, n0, lane, acc, bv);
  }
}

// ---------------------------------------------------------------------------
// Kernel 2: pointwise LSTM cell. 64*512 elements.
// ---------------------------------------------------------------------------
__global__ void __launch_bounds__(256)
lstm_cell_pointwise(const float* __restrict__ gates,
                    const float* __restrict__ c_in,
                    float* __restrict__ c_ws, float* __restrict__ h_ws,
                    float* __restrict__ cs_out /* d_out + B*T*V */, int t) {
  const int idx = blockIdx.x * blockDim.x + threadIdx.x;   // 0 .. 32767
  const int b = idx >> 9;
  const int j = idx & 511;
  const int P = Bsz * Hdim;

  float gi = gates[idx];
  float gf = gates[P + idx];
  float go = gates[2 * P + idx];
  float gz = gates[3 * P + idx];

  float i = 1.0f / (1.0f + __expf(-gi));
  float f = 1.0f / (1.0f + __expf(-gf));
  float o = 1.0f / (1.0f + __expf(-go));
  float z = tanhf(gz);

  float cn = i * z + f * c_in[idx];
  float hn = o * tanhf(cn);

  c_ws[idx] = cn;
  h_ws[idx] = hn;
  cs_out[((size_t)b * Tsz + t) * Hdim + j] = cn;
}

// ---------------------------------------------------------------------------
// Kernel 3: logits = h @ W_lin[lidx] + b_lin[lidx] -> straight into d_out slot.
// TDM streams W in 128x32 fp32 tiles (16KB) into double-buffered LDS; wave 0
// issues the DMA + s_wait_tensorcnt, workgroup barrier hands the buffer to all
// 8 waves, which consume B-fragments from LDS while the next tile is in
// flight. Each W_lin byte leaves HBM exactly once per step.
// grid = ceil(10000/32) = 313 blocks; wave = (mtile 0..3) x (nsub 0..1).
// ---------------------------------------------------------------------------
__global__ void __launch_bounds__(256)
logits_wmma(const float* __restrict__ h, const float* __restrict__ W,
            const float* __restrict__ bias, float* __restrict__ D, int ldd) {
  constexpr int NW = 32;    // columns per block
  constexpr int KT = 128;   // K rows per chunk
  __shared__ float wbuf[2][KT * NW];   // 2 x 16KB

  const int wave  = threadIdx.x >> 5;
  const int lane  = threadIdx.x & 31;
  const int m0    = (wave & 3) * 16;
  const int nsub  = wave >> 2;                  // 0..1
  const int nblk0 = blockIdx.x * NW;            // global column base
  const int ntile = blockIdx.x * 2 + nsub;
  const int n0    = ntile * 16;

  const unsigned int cols_rem = (unsigned int)(Vsz - nblk0); // OOB cols -> TDM zeros

#if defined(HAVE_TDM)
  if (wave == 0)
    tdm_load_2d((unsigned int)(uintptr_t)&wbuf[0][0], W + nblk0,
                cols_rem, (unsigned int)Hdim, (unsigned int)Vsz, NW, KT);
#endif

  v8f acc = {};
  for (int kc = 0; kc < Hdim; kc += KT) {
    const int bi = (kc / KT) & 1;
#if defined(HAVE_TDM)
    if (wave == 0) __builtin_amdgcn_s_wait_tensorcnt(0);
    __syncthreads();                      // buffer bi now valid for all waves
    if (wave == 0 && kc + KT < Hdim)      // prefetch next chunk into other buffer
      tdm_load_2d((unsigned int)(uintptr_t)&wbuf[bi ^ 1][0],
                  W + (size_t)(kc + KT) * Vsz + nblk0,
                  cols_rem, (unsigned int)(Hdim - (kc + KT)),
                  (unsigned int)Vsz, NW, KT);
#else
    for (int i = (int)threadIdx.x; i < KT * NW; i += 256) {
      int r = i / NW, c = i % NW;
      int gc = nblk0 + c;
      wbuf[bi][i] = (gc < Vsz) ? W[(size_t)(kc + r) * Vsz + gc] : 0.0f;
    }
    __syncthreads();
#endif
    const float* bufp = &wbuf[bi][0];
    #pragma unroll 8
    for (int k = 0; k < KT; k += 4) {
      v2f a = load_a(h, Hdim, m0, kc + k, lane);
      int n  = nsub * 16 + (lane & 15);
      int kk = k + ((lane >> 4) << 1);
      v2f b; b.x = bufp[kk * NW + n]; b.y = bufp[(kk + 1) * NW + n];
      acc = wmma4(a, b, acc);
    }
    __syncthreads();                      // all waves done with buffer bi
  }

  if (ntile < Vsz / 16)                   // wave-uniform edge guard (block 312)
    store_d(D, ldd, m0, n0, lane, acc, bias);
}

// ---------------------------------------------------------------------------
// Kernel 4: per-row softmax (in place in d_out), first-match argmax,
// embedding gather into x_next. One block per batch row.
// ---------------------------------------------------------------------------
__global__ void __launch_bounds__(256)
softmax_argmax_embed(const float* __restrict__ emb,
                     float* __restrict__ out,      // d_out base
                     float* __restrict__ x_next,   // [64,512]
                     int t) {
  const int b   = blockIdx.x;
  const int tid = threadIdx.x;
  float* L = out + ((size_t)b * Tsz + t) * Vsz;    // logits row (in place)

  __shared__ float sval[256];
  __shared__ int   sidx[256];
  __shared__ float ssum[256];

  float mv = -3.402823466e38f; int mi = 0;
  for (int j = tid; j < Vsz; j += 256) {
    float v = L[j];
    if (v > mv) { mv = v; mi = j; }
  }
  sval[tid] = mv; sidx[tid] = mi;
  __syncthreads();
  for (int s = 128; s > 0; s >>= 1) {
    if (tid < s) {
      float ov = sval[tid + s]; int oi = sidx[tid + s];
      if (ov > sval[tid] || (ov == sval[tid] && oi < sidx[tid])) {
        sval[tid] = ov; sidx[tid] = oi;
      }
    }
    __syncthreads();
  }
  const float M = sval[0];
  const int tok = sidx[0];

  float s = 0.0f;
  for (int j = tid; j < Vsz; j += 256) s += __expf(L[j] - M);
  ssum[tid] = s;
  __syncthreads();
  for (int st = 128; st > 0; st >>= 1) {
    if (tid < st) ssum[tid] += ssum[tid + st];
    __syncthreads();
  }
  const float inv = 1.0f / ssum[0];

  for (int j = tid; j < Vsz; j += 256) L[j] = __expf(L[j] - M) * inv;

  for (int j = tid; j < Xdim; j += 256)
    x_next[b * Xdim + j] = emb[(size_t)tok * Xdim + j];
}

// ---------------------------------------------------------------------------
extern "C" void kernel_launch(void* const* d_in, const int* in_sizes, int n_in,
                              void* d_out, int out_size, void* d_ws, size_t ws_size,
                              hipStream_t stream) {
  (void)in_sizes; (void)n_in; (void)out_size; (void)ws_size;

  const float* x  = (const float*)d_in[0];   // [64,32,512]
  const float* h0 = (const float*)d_in[1];   // [64,512]
  const float* c0 = (const float*)d_in[2];   // [64,512]

  GateParams gp;
  gp.Wh[0] = (const float*)d_in[3];  gp.Wx[0] = (const float*)d_in[4];  gp.bias[0] = (const float*)d_in[5];
  gp.Wh[1] = (const float*)d_in[6];  gp.Wx[1] = (const float*)d_in[7];  gp.bias[1] = (const float*)d_in[8];
  gp.Wh[2] = (const float*)d_in[9];  gp.Wx[2] = (const float*)d_in[10]; gp.bias[2] = (const float*)d_in[11];
  gp.Wh[3] = (const float*)d_in[12]; gp.Wx[3] = (const float*)d_in[13]; gp.bias[3] = (const float*)d_in[14];

  const float* W_lin = (const float*)d_in[15]; // [32,512,10000]
  const float* b_lin = (const float*)d_in[16]; // [32,10000]
  const float* emb   = (const float*)d_in[17]; // [10000,512]

  float* out    = (float*)d_out;
  float* cs_out = out + (size_t)Bsz * Tsz * Vsz;

  // workspace: gates[4*64*512] | h[64*512] | c[64*512] | x[64*512]  (~0.92 MB)
  float* ws    = (float*)d_ws;
  float* gates = ws;
  float* hbuf  = gates + 4 * Bsz * Hdim;
  float* cbuf  = hbuf + Bsz * Hdim;
  float* xbuf  = cbuf + Bsz * Hdim;

  for (int t = 0; t < Tsz; ++t) {
    const float* xA = (t == 0) ? x  : xbuf;   // x[:,0,:] row stride = T*X
    const int   ldx = (t == 0) ? Tsz * Xdim : Xdim;
    const float* hA = (t == 0) ? h0 : hbuf;
    const float* cA = (t == 0) ? c0 : cbuf;

    lstm_gates_wmma<<<32, 256, 0, stream>>>(xA, ldx, hA, Hdim, gp, gates);
    lstm_cell_pointwise<<<(Bsz * Hdim) / 256, 256, 0, stream>>>(
        gates, cA, cbuf, hbuf, cs_out, t);

    const int lidx = (t == 0) ? 0 : (t - 1);  // reference's linear_list indexing
    logits_wmma<<<(Vsz + 31) / 32, 256, 0, stream>>>(
        hbuf, W_lin + (size_t)lidx * Hdim * Vsz, b_lin + (size_t)lidx * Vsz,
        out + (size_t)t * Vsz, Tsz * Vsz);

    softmax_argmax_embed<<<Bsz, 256, 0, stream>>>(emb, out, xbuf, t);
  }
}